// Scalar_8675833938688
// MI455X (gfx1250) — compile-verified
//
#include <hip/hip_runtime.h>

// ---------------------------------------------------------------------------
// 2-D acoustic wave propagation with C-PML on MI455X (gfx1250).
// FD derivatives are computed as banded 16x16x20 matmuls on the WMMA pipe
// (V_WMMA_F32_16X16X4_F32, K chained in chunks of 4); pointwise PML updates
// run on VALU. One wave32 per 16x16 tile; one fused kernel per timestep.
// The banded FD operator matrices live in LDS (filled once per block with
// immediate stores), so A/B fragment fetches are ds_load_2addr pairs instead
// of per-matmul cndmask chains (which the register allocator kept
// rematerializing in earlier rounds).
// ---------------------------------------------------------------------------

typedef __attribute__((ext_vector_type(2))) float v2f;
typedef __attribute__((ext_vector_type(8))) float v8f;

namespace {
constexpr int   kNY = 256, kNX = 256, kPML = 20, kPAD = 22;
constexpr int   kNYP = kNY + 2 * kPAD;   // 300
constexpr int   kNXP = kNX + 2 * kPAD;   // 300
constexpr float kDY = 5.0f, kDX = 5.0f, kDT = 0.0005f;
constexpr int   kNT = 250, kNSHOT = 2, kNSRC = 8, kNREC = 64;
constexpr float kPMLFREQ = 25.0f, kMAXVEL = 4000.0f;
constexpr int   kTILES = (kNYP + 15) / 16;           // 19
constexpr int   kPlane = kNYP * kNXP;                // 90000
constexpr int   kField = kNSHOT * kPlane;            // 180000
}  // namespace

// D(16x16) += A(16x20) * B(20x16), as 5 chained V_WMMA_F32_16X16X4_F32.
// f32 fragment layout (ISA 7.12.2): A: lane&15 = M, lanes>=16 hold K+2 pair;
// B: lane&15 = N, lanes>=16 hold K+2 pair; C/D: VGPR j -> M = j + 8*(lane>=16).
template <typename FA, typename FB>
__device__ __forceinline__ v8f mm20(v8f acc, FA fa, FB fb, int hi) {
#pragma unroll
  for (int q = 0; q < 5; ++q) {
    const int k = 4 * q + 2 * hi;
    v2f a, b;
    a.x = fa(k);
    a.y = fa(k + 1);
    b.x = fb(k);
    b.y = fb(k + 1);
    acc = __builtin_amdgcn_wmma_f32_16x16x4_f32(false, a, false, b, (short)0,
                                                acc, false, false);
  }
  return acc;
}

__global__ __launch_bounds__(32) void step_kernel(
    const float* __restrict__ wfc, float* __restrict__ wf_other,  // wfm in / wfp out
    const float* __restrict__ psiy_in, float* __restrict__ psiy_out,
    const float* __restrict__ psix_in, float* __restrict__ psix_out,
    float* __restrict__ zetay, float* __restrict__ zetax,
    const float* __restrict__ v2dt2, const float* __restrict__ ay,
    const float* __restrict__ by, const float* __restrict__ ax,
    const float* __restrict__ bx) {
  __shared__ float sW[24][25];    // wfc window rows gy0-4..+19, cols gx0-4..+19
  __shared__ float sPyN[20][17];  // psiy_new, window rows gy0-2..gy0+17
  __shared__ float sPxN[16][21];  // psix_new, window cols gx0-2..gx0+17
  __shared__ float sOp1[20][17];  // banded C1/h operator:  sOp1[k][m] = C1[k-m]/h
  __shared__ float sOp2[20][17];  // banded C2/h^2 operator: sOp2[k][m] = C2[k-m]/h^2

  const int lane = threadIdx.x;
  const int hi = lane >> 4;    // K-pair / M-half selector
  const int lo = lane & 15;    // M (A-frags) or N (B-frags, C/D column)
  const int gy0 = blockIdx.y * 16;
  const int gx0 = blockIdx.x * 16;
  const size_t sb = (size_t)blockIdx.z * kPlane;

  // ---- Stage 0a: zero the operator tables (pure ds stores). ----
  for (int idx = lane; idx < 20 * 17; idx += 32) {
    (&sOp1[0][0])[idx] = 0.0f;
    (&sOp2[0][0])[idx] = 0.0f;
  }
  // ---- Stage 0b: write the 5 nonzero diagonals with immediate constants. --
  // DY == DX, so one table serves both the A-side operator (C1[k-m]) and the
  // transposed B-side operator (C1[k-n]).
  {
    const float i1 = 1.0f / kDY;          // == 1/kDX
    const float i2 = 1.0f / (kDY * kDY);  // == 1/(kDX*kDX)
    constexpr float C1c[5] = {1.0f / 12.0f, -2.0f / 3.0f, 0.0f, 2.0f / 3.0f,
                              -1.0f / 12.0f};
    constexpr float C2c[5] = {-1.0f / 12.0f, 4.0f / 3.0f, -5.0f / 2.0f,
                              4.0f / 3.0f, -1.0f / 12.0f};
    if (lane < 16) {
#pragma unroll
      for (int j = 0; j < 5; ++j) {
        sOp1[lane + j][lane] = C1c[j] * i1;
        sOp2[lane + j][lane] = C2c[j] * i2;
      }
    }
  }

  // ---- Stage 0c: stage the wfc window into LDS (zero Dirichlet halo). ----
  for (int idx = lane; idx < 24 * 24; idx += 32) {
    const int r = idx / 24, c = idx % 24;
    const int gy = gy0 - 4 + r, gx = gx0 - 4 + c;
    float v = 0.0f;
    if (gy >= 0 && gy < kNYP && gx >= 0 && gx < kNXP)
      v = wfc[sb + (size_t)gy * kNXP + gx];
    sW[r][c] = v;
  }
  __syncthreads();

  const v8f z = {};

  // ---- Stage 1: six 16x16x20 matmuls on the matrix pipe. ----
  // dwdy for grid rows gy0-2..gy0+13 and gy0+2..gy0+17:
  v8f Da = mm20(z, [&](int k) { return sOp1[k][lo]; },
                [&](int k) { return sW[k][4 + lo]; }, hi);
  v8f Db = mm20(z, [&](int k) { return sOp1[k][lo]; },
                [&](int k) { return sW[4 + k][4 + lo]; }, hi);
  // d2wdy2 on the tile:
  v8f De = mm20(z, [&](int k) { return sOp2[k][lo]; },
                [&](int k) { return sW[2 + k][4 + lo]; }, hi);
  // dwdx for grid cols gx0-2..gx0+13 and gx0+2..gx0+17:
  v8f Dc = mm20(z, [&](int k) { return sW[4 + lo][k]; },
                [&](int k) { return sOp1[k][lo]; }, hi);
  v8f Dd = mm20(z, [&](int k) { return sW[4 + lo][4 + k]; },
                [&](int k) { return sOp1[k][lo]; }, hi);
  // d2wdx2 on the tile:
  v8f Df = mm20(z, [&](int k) { return sW[4 + lo][2 + k]; },
                [&](int k) { return sOp2[k][lo]; }, hi);

  // ---- Stage 2: pointwise PML psi updates over the halo-extended windows. --
  {  // psiy_new: window rows w=0..19 <-> grid rows gy0-2+w, col gx0+lo.
    const int cx = gx0 + lo;
#pragma unroll
    for (int j = 0; j < 8; ++j) {
      const int w = j + 8 * hi;           // Da covers w = 0..15
      const int ry = gy0 - 2 + w;
      float pn = 0.0f;
      if (ry >= 0 && ry < kNYP && cx < kNXP) {
        const size_t ix = sb + (size_t)ry * kNXP + cx;
        pn = by[ry] * psiy_in[ix] + ay[ry] * Da[j];
      }
      sPyN[w][lo] = pn;
      if (w >= 2 && ry < kNYP && cx < kNXP)
        psiy_out[sb + (size_t)ry * kNXP + cx] = pn;
    }
#pragma unroll
    for (int j = 0; j < 8; ++j) {
      const int m = j + 8 * hi;
      if (m >= 12) {                      // Db m=12..15 covers w = 16..19
        const int w = m + 4;
        const int ry = gy0 - 2 + w;
        float pn = 0.0f;
        if (ry < kNYP && cx < kNXP) {
          const size_t ix = sb + (size_t)ry * kNXP + cx;
          pn = by[ry] * psiy_in[ix] + ay[ry] * Db[j];
        }
        sPyN[w][lo] = pn;
        if (w <= 17 && ry < kNYP && cx < kNXP)
          psiy_out[sb + (size_t)ry * kNXP + cx] = pn;
      }
    }
  }
  {  // psix_new: window cols u=0..19 <-> grid cols gx0-2+u, row gy0+m.
#pragma unroll
    for (int j = 0; j < 8; ++j) {
      const int m = j + 8 * hi;
      const int ry = gy0 + m;
      const int u = lo;                   // Dc covers u = 0..15
      const int cx = gx0 - 2 + u;
      float pn = 0.0f;
      if (cx >= 0 && cx < kNXP && ry < kNYP) {
        const size_t ix = sb + (size_t)ry * kNXP + cx;
        pn = bx[cx] * psix_in[ix] + ax[cx] * Dc[j];
      }
      sPxN[m][u] = pn;
      if (u >= 2 && ry < kNYP && cx < kNXP)
        psix_out[sb + (size_t)ry * kNXP + cx] = pn;
    }
    if (lo >= 12) {                       // Dd n=12..15 covers u = 16..19
#pragma unroll
      for (int j = 0; j < 8; ++j) {
        const int m = j + 8 * hi;
        const int ry = gy0 + m;
        const int u = lo + 4;
        const int cx = gx0 - 2 + u;
        float pn = 0.0f;
        if (cx < kNXP && ry < kNYP) {
          const size_t ix = sb + (size_t)ry * kNXP + cx;
          pn = bx[cx] * psix_in[ix] + ax[cx] * Dd[j];
        }
        sPxN[m][u] = pn;
        if (u <= 17 && ry < kNYP && cx < kNXP)
          psix_out[sb + (size_t)ry * kNXP + cx] = pn;
      }
    }
  }
  __syncthreads();

  // ---- Stage 3: dpsiydy / dpsixdx matmuls on the tile. ----
  v8f Dg = mm20(z, [&](int k) { return sOp1[k][lo]; },
                [&](int k) { return sPyN[k][lo]; }, hi);
  v8f Dh = mm20(z, [&](int k) { return sPxN[lo][k]; },
                [&](int k) { return sOp1[k][lo]; }, hi);

  // ---- Stage 4: zeta updates, Laplacian assembly, time update. ----
#pragma unroll
  for (int j = 0; j < 8; ++j) {
    const int m = j + 8 * hi;
    const int ry = gy0 + m, cx = gx0 + lo;
    if (ry < kNYP && cx < kNXP) {
      const size_t ix = sb + (size_t)ry * kNXP + cx;
      const float d2y = De[j], d2x = Df[j], dpy = Dg[j], dpx = Dh[j];
      const float zy = by[ry] * zetay[ix] + ay[ry] * (d2y + dpy);
      const float zxv = bx[cx] * zetax[ix] + ax[cx] * (d2x + dpx);
      const float lap = d2y + d2x + dpy + dpx + zy + zxv;
      const float wc = sW[4 + m][4 + lo];
      const float wp =
          2.0f * wc - wf_other[ix] + v2dt2[(size_t)ry * kNXP + cx] * lap;
      zetay[ix] = zy;
      zetax[ix] = zxv;
      wf_other[ix] = wp;  // wfp overwrites wfm in place (same-cell RAW only)
    }
  }
}

// Per-step source injection (atomic, duplicates accumulate as in .at[].add)
// then receiver gather of the just-updated wavefield.
__global__ void src_rec_kernel(float* __restrict__ wfp,
                               const float* __restrict__ v2dt2,
                               const int* __restrict__ src_loc,
                               const int* __restrict__ rec_loc,
                               const float* __restrict__ amps,
                               float* __restrict__ out, int t) {
  const int id = threadIdx.x;
  if (id < kNSHOT * kNSRC) {
    const int s = id / kNSRC;
    const int sy = src_loc[id * 2 + 0] + kPAD;
    const int sx = src_loc[id * 2 + 1] + kPAD;
    const float a = amps[id * kNT + t];
    const float scale = v2dt2[(size_t)sy * kNXP + sx];
    atomicAdd(&wfp[(size_t)s * kPlane + (size_t)sy * kNXP + sx], scale * a);
  }
  __syncthreads();
  if (id < kNSHOT * kNREC) {
    const int s = id / kNREC;
    const int ry = rec_loc[id * 2 + 0] + kPAD;
    const int rx = rec_loc[id * 2 + 1] + kPAD;
    out[id * kNT + t] = wfp[(size_t)s * kPlane + (size_t)ry * kNXP + rx];
  }
}

__global__ void zero_kernel(float* __restrict__ p, int n) {
  int i = blockIdx.x * blockDim.x + threadIdx.x;
  const int stride = gridDim.x * blockDim.x;
  for (; i < n; i += stride) p[i] = 0.0f;
}

__global__ void v2dt2_kernel(const float* __restrict__ v,
                             float* __restrict__ v2dt2) {
  const int i = blockIdx.x * blockDim.x + threadIdx.x;
  if (i >= kPlane) return;
  const int y = i / kNXP, x = i % kNXP;
  const int vy = min(max(y - kPAD, 0), kNY - 1);  // edge-replicate pad
  const int vx = min(max(x - kPAD, 0), kNX - 1);
  const float vv = v[vy * kNX + vx];
  v2dt2[i] = vv * vv * (kDT * kDT);
}

__device__ __forceinline__ void pml_prof(int idx, int n, float h, float* a,
                                         float* b) {
  const float lo = (float)kPAD, hif = (float)(n - kPAD - 1);
  float f = fmaxf(lo - (float)idx, (float)idx - hif) / (float)kPML;
  f = fminf(fmaxf(f, 0.0f), 1.0f);
  const float smax = 3.0f * kMAXVEL * 6.90775527898f / (2.0f * kPML * h);
  const float sigma = smax * f * f;
  const float alpha = 3.14159265358979f * kPMLFREQ * (1.0f - f);
  const float bb = __expf(-(sigma + alpha) * kDT);
  const float aa = sigma / (sigma + alpha + 1e-9f) * (bb - 1.0f);
  a[idx] = aa;
  b[idx] = bb;
}

__global__ void profiles_kernel(float* ay, float* by, float* ax, float* bx) {
  const int i = blockIdx.x * blockDim.x + threadIdx.x;
  if (i < kNYP) pml_prof(i, kNYP, kDY, ay, by);
  if (i >= kNYP && i < kNYP + kNXP) pml_prof(i - kNYP, kNXP, kDX, ax, bx);
}

extern "C" void kernel_launch(void* const* d_in, const int* in_sizes, int n_in,
                              void* d_out, int out_size, void* d_ws,
                              size_t ws_size, hipStream_t stream) {
  const float* v = (const float*)d_in[0];
  const float* amps = (const float*)d_in[1];
  const int* src_loc = (const int*)d_in[2];  // JAX default: int32 on device
  const int* rec_loc = (const int*)d_in[3];
  float* out = (float*)d_out;

  float* ws = (float*)d_ws;
  float* wf0 = ws;
  float* wf1 = wf0 + kField;
  float* py0 = wf1 + kField;
  float* py1 = py0 + kField;
  float* px0 = py1 + kField;
  float* px1 = px0 + kField;
  float* zy = px1 + kField;
  float* zx = zy + kField;
  float* v2 = zx + kField;
  float* ay = v2 + kPlane;
  float* by = ay + kNYP;
  float* ax = by + kNYP;
  float* bx = ax + kNXP;

  zero_kernel<<<256, 256, 0, stream>>>(ws, 8 * kField);
  v2dt2_kernel<<<(kPlane + 255) / 256, 256, 0, stream>>>(v, v2);
  profiles_kernel<<<(kNYP + kNXP + 255) / 256, 256, 0, stream>>>(ay, by, ax,
                                                                 bx);

  float *wc = wf0, *wo = wf1;
  float *pyi = py0, *pyo = py1;
  float *pxi = px0, *pxo = px1;
  const dim3 grid(kTILES, kTILES, kNSHOT);
  for (int t = 0; t < kNT; ++t) {
    step_kernel<<<grid, 32, 0, stream>>>(wc, wo, pyi, pyo, pxi, pxo, zy, zx,
                                         v2, ay, by, ax, bx);
    src_rec_kernel<<<1, 128, 0, stream>>>(wo, v2, src_loc, rec_loc, amps, out,
                                          t);
    float* tmp;
    tmp = wc; wc = wo; wo = tmp;
    tmp = pyi; pyi = pyo; pyo = tmp;
    tmp = pxi; pxi = pxo; pxo = tmp;
  }
  (void)in_sizes; (void)n_in; (void)out_size; (void)ws_size;
}